// GCN_67834713473299
// MI455X (gfx1250) — compile-verified
//
#include <hip/hip_runtime.h>
#include <hip/hip_bf16.h>

typedef float v2f __attribute__((ext_vector_type(2)));
typedef float v8f __attribute__((ext_vector_type(8)));

#define HCH 256
#define LDS_STRIDE 260   // 256 + 4 pad: keeps float4 alignment, spreads banks

// -------------------- degree (integer atomics) --------------------
__global__ __launch_bounds__(256) void deg_kernel(const int* __restrict__ row,
                                                  unsigned* __restrict__ deg, int nE) {
    int e = blockIdx.x * blockDim.x + threadIdx.x;
    if (e < nE) atomicAdd(&deg[row[e]], 1u);
}

// in-place: uint degree -> float rsqrt(1+deg)
__global__ __launch_bounds__(256) void norm_kernel(float* __restrict__ buf, int n) {
    int i = blockIdx.x * blockDim.x + threadIdx.x;
    if (i < n) {
        unsigned d = ((const unsigned*)buf)[i];
        buf[i] = rsqrtf(1.0f + (float)d);
    }
}

// -------------------- pos embedding: [N,119] @ [119,8] --------------------
__global__ __launch_bounds__(256) void pe_kernel(const float* __restrict__ pos_emb,
                                                 const float* __restrict__ lap_pe,
                                                 const float* __restrict__ Wp,
                                                 const float* __restrict__ bp,
                                                 float* __restrict__ pe, int n) {
    int i = blockIdx.x * blockDim.x + threadIdx.x;
    if (i >= n) return;
    float acc[8];
#pragma unroll
    for (int j = 0; j < 8; ++j) acc[j] = bp[j];
    for (int k = 0; k < 60; ++k) {
        float v = pos_emb[(size_t)i * 60 + k];
#pragma unroll
        for (int j = 0; j < 8; ++j) acc[j] += v * Wp[k * 8 + j];
    }
    for (int k = 0; k < 59; ++k) {
        float v = lap_pe[(size_t)i * 59 + k];
#pragma unroll
        for (int j = 0; j < 8; ++j) acc[j] += v * Wp[(60 + k) * 8 + j];
    }
#pragma unroll
    for (int j = 0; j < 8; ++j) pe[(size_t)i * 8 + j] = acc[j];
}

// -------------------- xemb GEMM via fp32 WMMA, fused epilogue --------------------
// g[i,:] = norm[i] * ( x[i,:] @ W[0:256,:] + pe[i,:] @ W[256:264,:] + bias )
// One block = 16 waves; block handles 16 rows; wave w handles cols [16w,16w+16).
__global__ __launch_bounds__(512) void xemb_wmma_kernel(const float* __restrict__ x,
                                                        const float* __restrict__ W,
                                                        const float* __restrict__ bias,
                                                        const float* __restrict__ pe,
                                                        const float* __restrict__ norm,
                                                        float* __restrict__ g, int nNodes) {
    __shared__ float xs[16 * LDS_STRIDE];

    const int tid = threadIdx.x;
    const int rowBase = blockIdx.x * 16;

    // stage 16x256 fp32 tile of x into LDS (float4, 2 per thread)
    for (int i = tid; i < 16 * 64; i += 512) {
        const int r = i >> 6, cq = i & 63;
        float4 v = make_float4(0.f, 0.f, 0.f, 0.f);
        if (rowBase + r < nNodes)
            v = ((const float4*)(x + (size_t)(rowBase + r) * HCH))[cq];
        float* dst = &xs[r * LDS_STRIDE + cq * 4];
        dst[0] = v.x; dst[1] = v.y; dst[2] = v.z; dst[3] = v.w;
    }
    __syncthreads();

    const int wave = tid >> 5;       // 0..15 -> column tile
    const int lane = tid & 31;
    const int half = lane >> 4;      // 0: K pair {k,k+1}; 1: {k+2,k+3}
    const int mn   = lane & 15;      // M for A frag / N for B frag
    const int nCol = wave * 16 + mn;

    v8f acc = {};
    for (int k0 = 0; k0 < HCH; k0 += 4) {
        const int ka = k0 + 2 * half;
        v2f a, b;
        a.x = xs[mn * LDS_STRIDE + ka];
        a.y = xs[mn * LDS_STRIDE + ka + 1];
        b.x = W[(size_t)ka * HCH + nCol];
        b.y = W[(size_t)(ka + 1) * HCH + nCol];
        acc = __builtin_amdgcn_wmma_f32_16x16x4_f32(false, a, false, b,
                                                    (short)0, acc, false, false);
    }

    const float bb = bias[nCol];
    float wpe[8];
#pragma unroll
    for (int j = 0; j < 8; ++j) wpe[j] = W[(size_t)(HCH + j) * HCH + nCol];

    // C layout: VGPR r, lanes 0-15 -> M=r; lanes 16-31 -> M=8+r; N = mn
#pragma unroll
    for (int r = 0; r < 8; ++r) {
        const int gRow = rowBase + r + 8 * half;
        if (gRow < nNodes) {
            float v = acc[r] + bb;
#pragma unroll
            for (int j = 0; j < 8; ++j) v += pe[(size_t)gRow * 8 + j] * wpe[j];
            g[(size_t)gRow * HCH + nCol] = norm[gRow] * v;
        }
    }
}

// -------------------- SpMM scatter-add: agg[row] += g[col] --------------------
// 64 threads per edge, float4 gather + 4 hardware f32 atomics each.
__global__ __launch_bounds__(256) void spmm_kernel(const float* __restrict__ g,
                                                   const int* __restrict__ row,
                                                   const int* __restrict__ col,
                                                   float* __restrict__ agg, int nE) {
    long gid = (long)blockIdx.x * blockDim.x + threadIdx.x;
    int e = (int)(gid >> 6);
    int t = (int)(gid & 63);
    if (e >= nE) return;
    const int src = col[e];
    const int dst = row[e];
    float4 v = ((const float4*)(g + (size_t)src * HCH))[t];
    float* o = agg + (size_t)dst * HCH + t * 4;
    unsafeAtomicAdd(o + 0, v.x);
    unsafeAtomicAdd(o + 1, v.y);
    unsafeAtomicAdd(o + 2, v.z);
    unsafeAtomicAdd(o + 3, v.w);
}

// -------------------- layer update, JK accumulation, next-layer scale --------------------
// h = norm*(agg + g);  out (=|+=) jk[layer]*h;  g = norm*h
__global__ __launch_bounds__(256) void update_kernel(const float* __restrict__ agg,
                                                     float* __restrict__ g,
                                                     const float* __restrict__ norm,
                                                     const float* __restrict__ jk,
                                                     int layer, int first,
                                                     float* __restrict__ out, int nNodes) {
    long i = (long)blockIdx.x * blockDim.x + threadIdx.x;   // float4 index
    long total = (long)nNodes * (HCH / 4);
    if (i >= total) return;
    const int node = (int)(i >> 6);                          // 64 float4 per node
    const float nm = norm[node];
    float4 a  = ((const float4*)agg)[i];
    float4 gg = ((const float4*)g)[i];
    float4 h;
    h.x = nm * (a.x + gg.x); h.y = nm * (a.y + gg.y);
    h.z = nm * (a.z + gg.z); h.w = nm * (a.w + gg.w);
    const float w = jk[layer];
    float4 o;
    if (first) {
        o.x = w * h.x; o.y = w * h.y; o.z = w * h.z; o.w = w * h.w;
    } else {
        float4 p = ((const float4*)out)[i];
        o.x = p.x + w * h.x; o.y = p.y + w * h.y;
        o.z = p.z + w * h.z; o.w = p.w + w * h.w;
    }
    ((float4*)out)[i] = o;
    float4 gn;
    gn.x = nm * h.x; gn.y = nm * h.y; gn.z = nm * h.z; gn.w = nm * h.w;
    ((float4*)g)[i] = gn;
}

extern "C" void kernel_launch(void* const* d_in, const int* in_sizes, int n_in,
                              void* d_out, int out_size, void* d_ws, size_t ws_size,
                              hipStream_t stream) {
    const float* x       = (const float*)d_in[0];
    const float* pos_emb = (const float*)d_in[1];
    const float* lap_pe  = (const float*)d_in[2];
    const float* W_pos   = (const float*)d_in[3];
    const float* b_pos   = (const float*)d_in[4];
    const float* W_xemb  = (const float*)d_in[5];
    const float* b_xemb  = (const float*)d_in[6];
    const float* jk      = (const float*)d_in[7];
    const int*   row     = (const int*)d_in[8];
    const int*   col     = (const int*)d_in[9];
    float* out = (float*)d_out;

    const int nNodes = in_sizes[0] / HCH;
    const int nE     = in_sizes[8];

    float* g     = (float*)d_ws;
    float* agg   = g + (size_t)nNodes * HCH;
    float* normb = agg + (size_t)nNodes * HCH;
    float* peb   = normb + nNodes;

    // degree -> norm
    hipMemsetAsync(normb, 0, (size_t)nNodes * sizeof(float), stream);
    deg_kernel<<<(nE + 255) / 256, 256, 0, stream>>>(row, (unsigned*)normb, nE);
    norm_kernel<<<(nNodes + 255) / 256, 256, 0, stream>>>(normb, nNodes);

    // pos embedding
    pe_kernel<<<(nNodes + 255) / 256, 256, 0, stream>>>(pos_emb, lap_pe, W_pos, b_pos, peb, nNodes);

    // fused concat-GEMM (WMMA fp32) + bias + norm scale -> g = norm*h0
    xemb_wmma_kernel<<<(nNodes + 15) / 16, 512, 0, stream>>>(x, W_xemb, b_xemb, peb, normb, g, nNodes);

    // 3 propagation layers
    for (int l = 0; l < 3; ++l) {
        hipMemsetAsync(agg, 0, (size_t)nNodes * HCH * sizeof(float), stream);
        long th = (long)nE * 64;
        spmm_kernel<<<(unsigned)((th + 255) / 256), 256, 0, stream>>>(g, row, col, agg, nE);
        long uth = (long)nNodes * (HCH / 4);
        update_kernel<<<(unsigned)((uth + 255) / 256), 256, 0, stream>>>(agg, g, normb, jk, l,
                                                                         l == 0 ? 1 : 0, out, nNodes);
    }
}